// RelationTransformer_40295383171629
// MI455X (gfx1250) — compile-verified
//
#include <hip/hip_runtime.h>
#include <hip/hip_bf16.h>

typedef float v2f __attribute__((ext_vector_type(2)));
typedef float v8f __attribute__((ext_vector_type(8)));

static __device__ __forceinline__ v8f wmma4(v2f a, v2f b, v8f c) {
  // D = A(16x4 f32) * B(4x16 f32) + C(16x16 f32), full fp32 precision
  return __builtin_amdgcn_wmma_f32_16x16x4_f32(false, a, false, b, (short)0, c,
                                               false, false);
}

static __device__ __forceinline__ void wave_reduce2(float& s, float& s2) {
  #pragma unroll
  for (int off = 16; off > 0; off >>= 1) {
    s  += __shfl_xor(s,  off, 32);
    s2 += __shfl_xor(s2, off, 32);
  }
}

// ---------------------------------------------------------------------------
// K1: P[b][m][c] = sum_k src[b][k][m] * wa1[k][c] + ba1[c]   (K=1024)
// src is (1028, 256) per batch => A is K-major (stride 256 between k).
// Per wave: 32x64 tile (2 A-frags x 4 B-frags = 8 WMMA / K-step, 12 loads).
// grid(8 batch, 4 rowblk of 64), block 256 (8 waves: 2 rowtiles x 4 colstrips)
// ---------------------------------------------------------------------------
__global__ void __launch_bounds__(256) k_gemm1(
    const float* __restrict__ tracks, const float* __restrict__ dets,
    const float* __restrict__ wa1, const float* __restrict__ ba1,
    float* __restrict__ P)
{
  const int batch = blockIdx.x;
  const int lane  = threadIdx.x & 31;
  const int wave  = threadIdx.x >> 5;
  const int m0 = blockIdx.y * 64 + (wave >> 2) * 32;
  const int c0 = (wave & 3) * 64;
  const int mn = lane & 15;          // row for A-frag, col for B/D-frag
  const int hi = lane >> 4;          // K half-select per ISA layout

  const float* src = (batch < 4) ? (tracks + (size_t)batch * 1028 * 256)
                                 : (dets + (size_t)(batch - 4) * 1028 * 256);
  const float* ap0 = src + m0 + mn;        // A[m][k]    = ap0[k*256]
  const float* ap1 = ap0 + 16;             // A[m+16][k]
  const float* bp  = wa1 + c0 + mn;        // B[k][c0+t*16+mn] = bp[k*256+t*16]

  v8f z = {0.f,0.f,0.f,0.f,0.f,0.f,0.f,0.f};
  v8f a00 = z, a01 = z, a02 = z, a03 = z;  // rows m0..m0+15
  v8f a10 = z, a11 = z, a12 = z, a13 = z;  // rows m0+16..m0+31

  #pragma unroll 2
  for (int k0 = 0; k0 < 1024; k0 += 4) {
    const int ka = k0 + hi * 2;
    v2f fa0; fa0.x = ap0[(size_t)ka * 256];      fa0.y = ap0[(size_t)(ka + 1) * 256];
    v2f fa1; fa1.x = ap1[(size_t)ka * 256];      fa1.y = ap1[(size_t)(ka + 1) * 256];
    v2f b0;  b0.x  = bp[(size_t)ka * 256 +  0];  b0.y  = bp[(size_t)(ka + 1) * 256 +  0];
    v2f b1;  b1.x  = bp[(size_t)ka * 256 + 16];  b1.y  = bp[(size_t)(ka + 1) * 256 + 16];
    v2f b2;  b2.x  = bp[(size_t)ka * 256 + 32];  b2.y  = bp[(size_t)(ka + 1) * 256 + 32];
    v2f b3;  b3.x  = bp[(size_t)ka * 256 + 48];  b3.y  = bp[(size_t)(ka + 1) * 256 + 48];
    a00 = wmma4(fa0, b0, a00);  a10 = wmma4(fa1, b0, a10);
    a01 = wmma4(fa0, b1, a01);  a11 = wmma4(fa1, b1, a11);
    a02 = wmma4(fa0, b2, a02);  a12 = wmma4(fa1, b2, a12);
    a03 = wmma4(fa0, b3, a03);  a13 = wmma4(fa1, b3, a13);
  }

  float* outb = P + (size_t)batch * 256 * 256;
  #pragma unroll
  for (int r = 0; r < 2; ++r) {
    #pragma unroll
    for (int t = 0; t < 4; ++t) {
      v8f acc = (r == 0) ? ((t == 0) ? a00 : (t == 1) ? a01 : (t == 2) ? a02 : a03)
                         : ((t == 0) ? a10 : (t == 1) ? a11 : (t == 2) ? a12 : a13);
      const int c = c0 + t * 16 + mn;
      const float bias = ba1[c];
      #pragma unroll
      for (int j = 0; j < 8; ++j) {
        const int row = m0 + 16 * r + j + 8 * hi;   // D: vgpr j = rows j / j+8
        outb[(size_t)row * 256 + c] = acc[j] + bias;
      }
    }
  }
}

// ---------------------------------------------------------------------------
// K2: per row r=b*256+n:
//   R[r][:] = relu(LN256(P[r][:]) * gla + bla)
//   G[r][:] = relu(LN128(geo[b][n][0..3] @ wg1 + bg1) * glg + blg)
// one wave per row; grid 256 blocks x 256 threads (8 rows / block)
// ---------------------------------------------------------------------------
__global__ void __launch_bounds__(256) k_lnrelu_g(
    const float* __restrict__ P,
    const float* __restrict__ tracks, const float* __restrict__ dets,
    const float* __restrict__ gla, const float* __restrict__ bla,
    const float* __restrict__ wg1, const float* __restrict__ bg1,
    const float* __restrict__ glg, const float* __restrict__ blg,
    float* __restrict__ R, float* __restrict__ G)
{
  const int lane = threadIdx.x & 31;
  const int row  = blockIdx.x * 8 + (threadIdx.x >> 5);
  const int b = row >> 8, n = row & 255;

  // ---- h path: LN over 256 then relu ----
  const float* prow = P + (size_t)row * 256;
  float x[8], s = 0.f, s2 = 0.f;
  #pragma unroll
  for (int i = 0; i < 8; ++i) {
    x[i] = prow[lane + 32 * i];
    s += x[i]; s2 += x[i] * x[i];
  }
  wave_reduce2(s, s2);
  float mean = s * (1.f / 256.f);
  float var  = s2 * (1.f / 256.f) - mean * mean;
  float rstd = rsqrtf(var + 1e-5f);
  #pragma unroll
  for (int i = 0; i < 8; ++i) {
    const int c = lane + 32 * i;
    float y = (x[i] - mean) * rstd * gla[c] + bla[c];
    R[(size_t)row * 256 + c] = fmaxf(y, 0.f);
  }

  // ---- g path: geo(4) @ wg1(4,128), LN over 128, relu ----
  const float* src = (b < 4) ? (tracks + (size_t)b * 1028 * 256)
                             : (dets + (size_t)(b - 4) * 1028 * 256);
  const float g0 = src[(size_t)1024 * 256 + n];
  const float g1 = src[(size_t)1025 * 256 + n];
  const float g2 = src[(size_t)1026 * 256 + n];
  const float g3 = src[(size_t)1027 * 256 + n];
  float u[4]; s = 0.f; s2 = 0.f;
  #pragma unroll
  for (int i = 0; i < 4; ++i) {
    const int c = lane + 32 * i;
    float uu = bg1[c];
    uu = fmaf(g0, wg1[c], uu);
    uu = fmaf(g1, wg1[128 + c], uu);
    uu = fmaf(g2, wg1[256 + c], uu);
    uu = fmaf(g3, wg1[384 + c], uu);
    u[i] = uu; s += uu; s2 += uu * uu;
  }
  wave_reduce2(s, s2);
  mean = s * (1.f / 128.f);
  var  = s2 * (1.f / 128.f) - mean * mean;
  rstd = rsqrtf(var + 1e-5f);
  #pragma unroll
  for (int i = 0; i < 4; ++i) {
    const int c = lane + 32 * i;
    float y = (u[i] - mean) * rstd * glg[c] + blg[c];
    G[(size_t)row * 128 + c] = fmaxf(y, 0.f);
  }
}

// ---------------------------------------------------------------------------
// K3: EPRE[r][c] = R[r][:]@wa2 + G[r][:]@wg2 + ba2[c] + bg2[c]
// A row-major (packed v2f loads). Per wave 32x64.
// grid(8 batch, 4 rowblk of 64), block 256
// ---------------------------------------------------------------------------
__global__ void __launch_bounds__(256) k_gemm2(
    const float* __restrict__ R, const float* __restrict__ G,
    const float* __restrict__ wa2, const float* __restrict__ wg2,
    const float* __restrict__ ba2, const float* __restrict__ bg2,
    float* __restrict__ EPRE)
{
  const int batch = blockIdx.x;
  const int lane  = threadIdx.x & 31;
  const int wave  = threadIdx.x >> 5;
  const int m0 = blockIdx.y * 64 + (wave >> 2) * 32;
  const int c0 = (wave & 3) * 64;
  const int mn = lane & 15;
  const int hi = lane >> 4;

  const float* ar0 = R + ((size_t)batch * 256 + m0 + mn) * 256;        // contiguous k
  const float* ar1 = ar0 + (size_t)16 * 256;
  const float* ag0 = G + ((size_t)batch * 256 + m0 + mn) * 128;
  const float* ag1 = ag0 + (size_t)16 * 128;
  const float* bp1 = wa2 + c0 + mn;
  const float* bp2 = wg2 + c0 + mn;

  v8f z = {0.f,0.f,0.f,0.f,0.f,0.f,0.f,0.f};
  v8f a00 = z, a01 = z, a02 = z, a03 = z;
  v8f a10 = z, a11 = z, a12 = z, a13 = z;

  #pragma unroll 2
  for (int k0 = 0; k0 < 256; k0 += 4) {
    const int ka = k0 + hi * 2;
    v2f fa0 = *(const v2f*)(ar0 + ka);
    v2f fa1 = *(const v2f*)(ar1 + ka);
    v2f b0; b0.x = bp1[(size_t)ka * 256 +  0]; b0.y = bp1[(size_t)(ka + 1) * 256 +  0];
    v2f b1; b1.x = bp1[(size_t)ka * 256 + 16]; b1.y = bp1[(size_t)(ka + 1) * 256 + 16];
    v2f b2; b2.x = bp1[(size_t)ka * 256 + 32]; b2.y = bp1[(size_t)(ka + 1) * 256 + 32];
    v2f b3; b3.x = bp1[(size_t)ka * 256 + 48]; b3.y = bp1[(size_t)(ka + 1) * 256 + 48];
    a00 = wmma4(fa0, b0, a00);  a10 = wmma4(fa1, b0, a10);
    a01 = wmma4(fa0, b1, a01);  a11 = wmma4(fa1, b1, a11);
    a02 = wmma4(fa0, b2, a02);  a12 = wmma4(fa1, b2, a12);
    a03 = wmma4(fa0, b3, a03);  a13 = wmma4(fa1, b3, a13);
  }
  #pragma unroll 2
  for (int k0 = 0; k0 < 128; k0 += 4) {
    const int ka = k0 + hi * 2;
    v2f fa0 = *(const v2f*)(ag0 + ka);
    v2f fa1 = *(const v2f*)(ag1 + ka);
    v2f b0; b0.x = bp2[(size_t)ka * 256 +  0]; b0.y = bp2[(size_t)(ka + 1) * 256 +  0];
    v2f b1; b1.x = bp2[(size_t)ka * 256 + 16]; b1.y = bp2[(size_t)(ka + 1) * 256 + 16];
    v2f b2; b2.x = bp2[(size_t)ka * 256 + 32]; b2.y = bp2[(size_t)(ka + 1) * 256 + 32];
    v2f b3; b3.x = bp2[(size_t)ka * 256 + 48]; b3.y = bp2[(size_t)(ka + 1) * 256 + 48];
    a00 = wmma4(fa0, b0, a00);  a10 = wmma4(fa1, b0, a10);
    a01 = wmma4(fa0, b1, a01);  a11 = wmma4(fa1, b1, a11);
    a02 = wmma4(fa0, b2, a02);  a12 = wmma4(fa1, b2, a12);
    a03 = wmma4(fa0, b3, a03);  a13 = wmma4(fa1, b3, a13);
  }

  float* outb = EPRE + (size_t)batch * 256 * 256;
  #pragma unroll
  for (int r = 0; r < 2; ++r) {
    #pragma unroll
    for (int t = 0; t < 4; ++t) {
      v8f acc = (r == 0) ? ((t == 0) ? a00 : (t == 1) ? a01 : (t == 2) ? a02 : a03)
                         : ((t == 0) ? a10 : (t == 1) ? a11 : (t == 2) ? a12 : a13);
      const int c = c0 + t * 16 + mn;
      const float bias = ba2[c] + bg2[c];
      #pragma unroll
      for (int j = 0; j < 8; ++j) {
        const int row = m0 + 16 * r + j + 8 * hi;
        outb[(size_t)row * 256 + c] = acc[j] + bias;
      }
    }
  }
}

// ---------------------------------------------------------------------------
// K4: EMB[r][:] = LN256(EPRE[r][:]) * gf + bf   (one wave per row)
// ---------------------------------------------------------------------------
__global__ void __launch_bounds__(256) k_finln(
    const float* __restrict__ EPRE,
    const float* __restrict__ gf, const float* __restrict__ bf,
    float* __restrict__ EMB)
{
  const int lane = threadIdx.x & 31;
  const int row  = blockIdx.x * 8 + (threadIdx.x >> 5);
  const float* prow = EPRE + (size_t)row * 256;
  float x[8], s = 0.f, s2 = 0.f;
  #pragma unroll
  for (int i = 0; i < 8; ++i) {
    x[i] = prow[lane + 32 * i];
    s += x[i]; s2 += x[i] * x[i];
  }
  wave_reduce2(s, s2);
  const float mean = s * (1.f / 256.f);
  const float var  = s2 * (1.f / 256.f) - mean * mean;
  const float rstd = rsqrtf(var + 1e-5f);
  #pragma unroll
  for (int i = 0; i < 8; ++i) {
    const int c = lane + 32 * i;
    EMB[(size_t)row * 256 + c] = (x[i] - mean) * rstd * gf[c] + bf[c];
  }
}

// ---------------------------------------------------------------------------
// K5: AC[b][m][0..127] = EMB[b][m][:] @ wc1(top half for b<4, bottom for b>=4)
// Per wave 32x64. grid(8 batch, 2 rowblk of 128), block 256
// (8 waves: 4 rowtiles x 2 colstrips)
// ---------------------------------------------------------------------------
__global__ void __launch_bounds__(256) k_gemm3(
    const float* __restrict__ EMB, const float* __restrict__ wc1,
    float* __restrict__ AC)
{
  const int batch = blockIdx.x;
  const int lane  = threadIdx.x & 31;
  const int wave  = threadIdx.x >> 5;
  const int m0 = blockIdx.y * 128 + (wave >> 1) * 32;
  const int c0 = (wave & 1) * 64;
  const int mn = lane & 15;
  const int hi = lane >> 4;

  const float* ar0 = EMB + ((size_t)batch * 256 + m0 + mn) * 256;
  const float* ar1 = ar0 + (size_t)16 * 256;
  const float* w   = wc1 + ((batch >= 4) ? (size_t)256 * 128 : 0);
  const float* bp  = w + c0 + mn;

  v8f z = {0.f,0.f,0.f,0.f,0.f,0.f,0.f,0.f};
  v8f a00 = z, a01 = z, a02 = z, a03 = z;
  v8f a10 = z, a11 = z, a12 = z, a13 = z;

  #pragma unroll 2
  for (int k0 = 0; k0 < 256; k0 += 4) {
    const int ka = k0 + hi * 2;
    v2f fa0 = *(const v2f*)(ar0 + ka);
    v2f fa1 = *(const v2f*)(ar1 + ka);
    v2f b0; b0.x = bp[(size_t)ka * 128 +  0]; b0.y = bp[(size_t)(ka + 1) * 128 +  0];
    v2f b1; b1.x = bp[(size_t)ka * 128 + 16]; b1.y = bp[(size_t)(ka + 1) * 128 + 16];
    v2f b2; b2.x = bp[(size_t)ka * 128 + 32]; b2.y = bp[(size_t)(ka + 1) * 128 + 32];
    v2f b3; b3.x = bp[(size_t)ka * 128 + 48]; b3.y = bp[(size_t)(ka + 1) * 128 + 48];
    a00 = wmma4(fa0, b0, a00);  a10 = wmma4(fa1, b0, a10);
    a01 = wmma4(fa0, b1, a01);  a11 = wmma4(fa1, b1, a11);
    a02 = wmma4(fa0, b2, a02);  a12 = wmma4(fa1, b2, a12);
    a03 = wmma4(fa0, b3, a03);  a13 = wmma4(fa1, b3, a13);
  }

  float* outb = AC + (size_t)batch * 256 * 128;
  #pragma unroll
  for (int r = 0; r < 2; ++r) {
    #pragma unroll
    for (int t = 0; t < 4; ++t) {
      v8f acc = (r == 0) ? ((t == 0) ? a00 : (t == 1) ? a01 : (t == 2) ? a02 : a03)
                         : ((t == 0) ? a10 : (t == 1) ? a11 : (t == 2) ? a12 : a13);
      const int c = c0 + t * 16 + mn;
      #pragma unroll
      for (int j = 0; j < 8; ++j) {
        const int row = m0 + 16 * r + j + 8 * hi;
        outb[(size_t)row * 128 + c] = acc[j];
      }
    }
  }
}

// ---------------------------------------------------------------------------
// K6: logits[b][n][m] = bc2 + sum_h relu(a[b][n][h] + c[b][m][h] + bc1[h])*wc2[h]
// grid(4, 8, 8): 32x32 (n,m) tile per block; LDS-staged with pad 132 vs 64-bank
// ---------------------------------------------------------------------------
__global__ void __launch_bounds__(256) k_pair(
    const float* __restrict__ AC,
    const float* __restrict__ bc1, const float* __restrict__ wc2,
    const float* __restrict__ bc2, float* __restrict__ out)
{
  __shared__ float aT[32][132];
  __shared__ float cT[32][132];
  __shared__ float w2[128];

  const int b = blockIdx.x, nblk = blockIdx.y, mblk = blockIdx.z;
  const float* arow = AC + ((size_t)b * 256 + nblk * 32) * 128;
  const float* crow = AC + ((size_t)(b + 4) * 256 + mblk * 32) * 128;

  for (int i = threadIdx.x; i < 32 * 128; i += 256) {
    const int r = i >> 7, h = i & 127;
    aT[r][h] = arow[(size_t)r * 128 + h] + bc1[h];  // fold bc1 into a-tile
    cT[r][h] = crow[(size_t)r * 128 + h];
  }
  if (threadIdx.x < 128) w2[threadIdx.x] = wc2[threadIdx.x];
  __syncthreads();

  const float b2 = bc2[0];
  const int n = threadIdx.x >> 3;
  const int mbase = (threadIdx.x & 7) * 4;

  float acc[4] = {0.f, 0.f, 0.f, 0.f};
  #pragma unroll 4
  for (int h = 0; h < 128; ++h) {
    const float av = aT[n][h];
    const float wv = w2[h];
    #pragma unroll
    for (int j = 0; j < 4; ++j)
      acc[j] = fmaf(fmaxf(av + cT[mbase + j][h], 0.f), wv, acc[j]);
  }

  float* orow = out + ((size_t)b * 256 + nblk * 32 + n) * 256 + mblk * 32 + mbase;
  #pragma unroll
  for (int j = 0; j < 4; ++j) orow[j] = acc[j] + b2;
}

// ---------------------------------------------------------------------------
extern "C" void kernel_launch(void* const* d_in, const int* in_sizes, int n_in,
                              void* d_out, int out_size, void* d_ws, size_t ws_size,
                              hipStream_t stream) {
  const float* tracks = (const float*)d_in[0];
  const float* dets   = (const float*)d_in[1];
  const float* wa1 = (const float*)d_in[2];
  const float* ba1 = (const float*)d_in[3];
  const float* gla = (const float*)d_in[4];
  const float* bla = (const float*)d_in[5];
  const float* wa2 = (const float*)d_in[6];
  const float* ba2 = (const float*)d_in[7];
  const float* wg1 = (const float*)d_in[8];
  const float* bg1 = (const float*)d_in[9];
  const float* glg = (const float*)d_in[10];
  const float* blg = (const float*)d_in[11];
  const float* wg2 = (const float*)d_in[12];
  const float* bg2 = (const float*)d_in[13];
  const float* gf  = (const float*)d_in[14];
  const float* bf  = (const float*)d_in[15];
  // d_in[16..23]: wq,bq,wk,bk,wv,bv,wo,bo — dead in the reference (attn_out unused)
  const float* wc1 = (const float*)d_in[24];
  const float* bc1 = (const float*)d_in[25];
  const float* wc2 = (const float*)d_in[26];
  const float* bc2 = (const float*)d_in[27];
  float* logits = (float*)d_out;

  // workspace layout (bytes), with reuse across pipeline stages
  char* ws = (char*)d_ws;
  float* P    = (float*)(ws + 0);                    // 8*256*256 = 2 MB
  float* R    = (float*)(ws + 2 * 1024 * 1024);      // 2 MB
  float* G    = (float*)(ws + 4 * 1024 * 1024);      // 1 MB
  float* EPRE = P;     // P dead after k_lnrelu_g
  float* EMB  = R;     // R dead after k_gemm2
  float* AC   = G;     // G dead after k_gemm2 (AC needs 1 MB)

  k_gemm1<<<dim3(8, 4), 256, 0, stream>>>(tracks, dets, wa1, ba1, P);
  k_lnrelu_g<<<dim3(256), 256, 0, stream>>>(P, tracks, dets, gla, bla,
                                            wg1, bg1, glg, blg, R, G);
  k_gemm2<<<dim3(8, 4), 256, 0, stream>>>(R, G, wa2, wg2, ba2, bg2, EPRE);
  k_finln<<<dim3(256), 256, 0, stream>>>(EPRE, gf, bf, EMB);
  k_gemm3<<<dim3(8, 2), 256, 0, stream>>>(EMB, wc1, AC);
  k_pair<<<dim3(4, 8, 8), 256, 0, stream>>>(AC, bc1, wc2, bc2, logits);
}